// MAMCLoss_64080912056617
// MI455X (gfx1250) — compile-verified
//
#include <hip/hip_runtime.h>
#include <hip/hip_bf16.h>
#include <math.h>
#include <stdint.h>

typedef __attribute__((ext_vector_type(2))) float v2f;
typedef __attribute__((ext_vector_type(8))) float v8f;

namespace {
constexpr int   kBatch   = 2048;
constexpr int   kCls     = 200;
constexpr int   kDim     = 512;
constexpr int   kRows    = 8192;                 // 2048 * 4 parts
constexpr int   kStride  = 516;                  // padded LDS pitch (floats): conflict-free
constexpr int   kStrip   = 64;                   // i-rows per workgroup (4 tiles, j-split x2)
constexpr int   kBRows   = 32;                   // B rows staged per round
constexpr int   kRounds  = kRows / kBRows;       // 256
constexpr float kLambda  = 0.5f;
constexpr float kEps     = 0.1f;
constexpr int   kAFloats = kStrip * kStride;     // 33024
constexpr int   kBHalf   = kBRows * kStride;     // 16512 (one ping/pong buffer)
constexpr int   kShFloats = kAFloats + 2 * kBHalf + kBatch;  // 68096 -> ~272 KB
}

// --- CDNA5 async global->LDS copy (bypasses VGPR staging, tracked by ASYNCcnt) ---
__device__ __forceinline__ void async_b128(unsigned lds_byte_off, const void* gaddr) {
  asm volatile("global_load_async_to_lds_b128 %0, %1, off"
               :: "v"(lds_byte_off), "v"(gaddr)
               : "memory");
}
// ASYNCcnt decrements in order -> waiting <=16 means "all but the newest batch done"
__device__ __forceinline__ void wait_async16() {
  asm volatile("s_wait_asynccnt 0x10" ::: "memory");
}
__device__ __forceinline__ void wait_async0() {
  asm volatile("s_wait_asynccnt 0x0" ::: "memory");
}
__device__ __forceinline__ unsigned lds_off(const void* p) {
  return (unsigned)(uintptr_t)p;   // LDS aperture: bits [31:0] are the LDS byte offset
}

__global__ void mamc_zero(float* out) { if (threadIdx.x == 0) out[0] = 0.0f; }
__global__ void mamc_zero_ws(float* ws) { ws[blockIdx.x * 256 + threadIdx.x] = 0.0f; }

// Row L2-normalize: one block (128 threads) per row, float4 per lane.
__global__ void mamc_normalize(const float* __restrict__ x, float* __restrict__ f)
{
  const int row = blockIdx.x;
  const int t   = threadIdx.x;
  const float4 v = reinterpret_cast<const float4*>(x + (size_t)row * kDim)[t];
  float ss = v.x*v.x + v.y*v.y + v.z*v.z + v.w*v.w;
  #pragma unroll
  for (int off = 16; off >= 1; off >>= 1) ss += __shfl_xor(ss, off, 32);
  __shared__ float ws[4];
  if ((t & 31) == 0) ws[t >> 5] = ss;
  __syncthreads();
  const float tot = ws[0] + ws[1] + ws[2] + ws[3];
  const float s = 1.0f / fmaxf(sqrtf(tot), 1e-12f);
  float4 o; o.x = v.x*s; o.y = v.y*s; o.z = v.z*s; o.w = v.w*s;
  reinterpret_cast<float4*>(f + (size_t)row * kDim)[t] = o;
}

// Label-smoothed cross entropy: one wave per row.
__global__ void mamc_ce(const float* __restrict__ pred, const int* __restrict__ tgt,
                        float* __restrict__ out)
{
  const int lane = threadIdx.x & 31;
  const int row  = blockIdx.x * 8 + (threadIdx.x >> 5);
  const float* p = pred + (size_t)row * kCls;
  float mx = -3.402823466e38f;
  for (int c = lane; c < kCls; c += 32) mx = fmaxf(mx, p[c]);
  #pragma unroll
  for (int off = 16; off >= 1; off >>= 1) mx = fmaxf(mx, __shfl_xor(mx, off, 32));
  float se = 0.f, sm = 0.f;
  for (int c = lane; c < kCls; c += 32) { const float v = p[c]; se += __expf(v - mx); sm += v; }
  #pragma unroll
  for (int off = 16; off >= 1; off >>= 1) { se += __shfl_xor(se, off, 32); sm += __shfl_xor(sm, off, 32); }
  if (lane == 0) {
    const float lse    = mx + __logf(se);
    const float nll    = lse - p[tgt[row]];
    const float smooth = lse - sm * (1.0f / kCls);
    atomicAdd(out, ((1.0f - kEps) * nll + kEps * smooth) * (1.0f / kBatch));
  }
}

// PASS 0: E_A[i] += sum_{!(sc&&sa)} exp(prod), E_B[i] += sum_{!sc&&!sa} exp(prod)
// PASS 1: out += lambda/n * sum over positive masks of log1p(exp(-prod) * E)
template<int PASS>
__global__ __launch_bounds__(256)
void mamc_npairs(const float* __restrict__ f, const int* __restrict__ tgt,
                 float* __restrict__ EA, float* __restrict__ EB,
                 float* __restrict__ out)
{
  extern __shared__ float lds[];
  float* As  = lds;                                   // 64 x 512 (pitch 516)
  float* Bs0 = lds + kAFloats;                        // ping: 32 x 512 (pitch 516)
  float* Bs1 = Bs0 + kBHalf;                          // pong
  int*   Ts  = reinterpret_cast<int*>(Bs1 + kBHalf);  // 2048 targets

  const int tid   = threadIdx.x;
  const int wave  = tid >> 5;
  const int lane  = tid & 31;
  const int half  = lane >> 4;          // K-pair select in A/B frags; M+8 in C
  const int lrow  = lane & 15;
  const int iBase = blockIdx.x * kStrip;
  const int iTile = iBase + (wave & 3) * 16;   // waves w and w+4 share an i-tile
  const int jSel  = (wave >> 2) * 16;          // ...and split each 32-row B round

  // Coalesced chunk map: chunk c = tid + 256*u  ->  row=(tid>>7)+2u, col=(tid&127)*4
  const int myRow = tid >> 7;                  // 0 or 1
  const int myCol = (tid & 127) << 2;

  // ---- Prologue: async-stage targets + A strip, drain, barrier ----
  #pragma unroll
  for (int u = 0; u < 2; ++u) {
    const int c = tid + 256 * u;               // c in [0, 512)
    async_b128(lds_off(Ts + c * 4), tgt + c * 4);
  }
  {
    const unsigned aBase = lds_off(As + myRow * kStride + myCol);
    const float*   gBase = f + (size_t)(iBase + myRow) * kDim + myCol;
    #pragma unroll
    for (int u = 0; u < 32; ++u)               // 64 rows x 128 chunks / 256 thr
      async_b128(aBase + u * (2 * kStride * 4), gBase + (size_t)(2 * u) * kDim);
  }
  wait_async0();
  __syncthreads();

  int   tI[8], pI[8];
  float eAi[8], eBi[8];
  #pragma unroll
  for (int r = 0; r < 8; ++r) {
    const int i = iTile + r + 8 * half;        // C/D layout: VGPR r, lanes>=16 -> M+8
    tI[r] = Ts[i >> 2];
    pI[r] = i & 3;
    if (PASS == 1) { eAi[r] = EA[i]; eBi[r] = EB[i]; }
  }

  float accA[8] = {0,0,0,0,0,0,0,0};
  float accB[8] = {0,0,0,0,0,0,0,0};
  float lsum = 0.f;

  const float* ap = As + ((wave & 3) * 16 + lrow) * kStride + 2 * half;
  const unsigned bBase[2] = { lds_off(Bs0 + myRow * kStride + myCol),
                              lds_off(Bs1 + myRow * kStride + myCol) };
  const float* gB = f + (size_t)myRow * kDim + myCol;   // + (32*r)*kDim per round

  // Issue round 0 into ping
  #pragma unroll
  for (int u = 0; u < 16; ++u)                 // 32 rows x 128 chunks / 256 thr
    async_b128(bBase[0] + u * (2 * kStride * 4), gB + (size_t)(2 * u) * kDim);

  for (int rnd = 0; rnd < kRounds; ++rnd) {
    const int cur = rnd & 1;
    if (rnd + 1 < kRounds) {                   // prefetch next round into other buffer
      const float* g = gB + (size_t)(rnd + 1) * kBRows * kDim;
      #pragma unroll
      for (int u = 0; u < 16; ++u)
        async_b128(bBase[cur ^ 1] + u * (2 * kStride * 4), g + (size_t)(2 * u) * kDim);
      wait_async16();                          // current round's batch complete
    } else {
      wait_async0();
    }
    __syncthreads();                           // all waves see staged data

    const float* bp = (cur ? Bs1 : Bs0) + (jSel + lrow) * kStride + 2 * half;
    v8f acc = {};
    #pragma unroll 8
    for (int k = 0; k < kDim; k += 4) {
      const v2f a = *reinterpret_cast<const v2f*>(ap + k);
      const v2f b = *reinterpret_cast<const v2f*>(bp + k);
      acc = __builtin_amdgcn_wmma_f32_16x16x4_f32(
          false, a, false, b, (short)0, acc, false, false);
    }
    __syncthreads();                           // buffer consumed: next round may overwrite

    const int j  = rnd * kBRows + jSel + lrow; // this lane's column
    const int tJ = Ts[j >> 2];
    const int pJ = j & 3;
    #pragma unroll
    for (int r = 0; r < 8; ++r) {
      const float prod = acc[r];
      const bool sc = (tI[r] == tJ);
      const bool sa = (pI[r] == pJ);
      if (PASS == 0) {
        const float e = __expf(prod);          // |prod| <= 1: fast exp is safe
        accA[r] += (sc && sa) ? 0.f : e;
        accB[r] += (sc || sa) ? 0.f : e;
      } else {
        const float einv = __expf(-prod);
        const float E    = (sc && sa) ? eAi[r] : eBi[r];
        if (sc || sa) lsum += __logf(fmaf(einv, E, 1.0f));  // log1p, arg >> eps
      }
    }
  }

  if (PASS == 0) {
    #pragma unroll
    for (int r = 0; r < 8; ++r) {
      float a = accA[r], b = accB[r];
      #pragma unroll
      for (int off = 1; off < 16; off <<= 1) {
        a += __shfl_xor(a, off, 32);
        b += __shfl_xor(b, off, 32);
      }
      if (lrow == 0) {                         // two waves per row -> atomic combine
        const int i = iTile + r + 8 * half;
        atomicAdd(&EA[i], a);
        atomicAdd(&EB[i], b);
      }
    }
  } else {
    #pragma unroll
    for (int off = 16; off >= 1; off >>= 1) lsum += __shfl_xor(lsum, off, 32);
    if (lane == 0) atomicAdd(out, lsum * (kLambda / (float)kRows));
  }
}

extern "C" void kernel_launch(void* const* d_in, const int* in_sizes, int n_in,
                              void* d_out, int out_size, void* d_ws, size_t ws_size,
                              hipStream_t stream) {
  (void)in_sizes; (void)n_in; (void)out_size; (void)ws_size;
  const float* pred  = (const float*)d_in[0];
  const float* xpart = (const float*)d_in[1];
  const int*   tgt   = (const int*)d_in[2];
  float* out = (float*)d_out;

  float* fnorm = (float*)d_ws;                         // 8192*512 f32 = 16 MB
  float* EA    = fnorm + (size_t)kRows * kDim;         // 8192 f32
  float* EB    = EA + kRows;                           // 8192 f32

  mamc_zero<<<1, 32, 0, stream>>>(out);
  mamc_zero_ws<<<64, 256, 0, stream>>>(EA);            // zeros EA and EB (contiguous 16384)
  mamc_normalize<<<kRows, kDim / 4, 0, stream>>>(xpart, fnorm);
  mamc_ce<<<kBatch / 8, 256, 0, stream>>>(pred, tgt, out);

  const size_t shbytes = (size_t)kShFloats * sizeof(float);  // ~272 KB < 320 KB WGP LDS
  mamc_npairs<0><<<kRows / kStrip, 256, shbytes, stream>>>(fnorm, tgt, EA, EB, out);
  mamc_npairs<1><<<kRows / kStrip, 256, shbytes, stream>>>(fnorm, tgt, EA, EB, out);
}